// GCN_4432406250065
// MI455X (gfx1250) — compile-verified
//
#include <hip/hip_runtime.h>

typedef __attribute__((ext_vector_type(2))) float v2f;
typedef __attribute__((ext_vector_type(8))) float v8f;

#define LDSS 132   // 16-row A tile stride in floats; 132%64=4 -> conflict-free column reads

// ---------------- utility kernels ----------------

__global__ __launch_bounds__(256) void gcn_zero(float* __restrict__ p, long n) {
    long i = (long)blockIdx.x * blockDim.x + threadIdx.x;
    long stride = (long)gridDim.x * blockDim.x;
    for (; i < n; i += stride) p[i] = 0.0f;
}

__global__ __launch_bounds__(256) void gcn_degree(const int* __restrict__ src,
                                                  const int* __restrict__ dst,
                                                  float* __restrict__ deg_src,
                                                  float* __restrict__ deg_dst, int E) {
    int e = blockIdx.x * blockDim.x + threadIdx.x;
    if (e < E) {
        atomicAdd(&deg_src[src[e]], 1.0f);
        atomicAdd(&deg_dst[dst[e]], 1.0f);
    }
}

__global__ __launch_bounds__(256) void gcn_norm(float* __restrict__ deg, int n) {
    int i = blockIdx.x * blockDim.x + threadIdx.x;
    if (i < n) deg[i] = rsqrtf(fmaxf(deg[i], 1.0f));   // clamp deg>=1 like DGL
}

// ---------------- layer 1 GEMM: H1 = (feats * norm_src) @ W1  [N,128]x[128,128] ----------------
// 256 threads = 8 waves; block handles 16 rows x 128 cols (wave w -> col tile 16w).

__global__ __launch_bounds__(256) void gcn_gemm1(const float* __restrict__ feats,
                                                 const float* __restrict__ norm_src,
                                                 const float* __restrict__ W1,
                                                 float* __restrict__ H1, int nn) {
    __shared__ float As[16 * LDSS];
    const int tid  = threadIdx.x;
    const int wave = tid >> 5;
    const int lane = tid & 31;
    const int mbase = blockIdx.x * 16;

    // stage A tile (scaled by norm_src) into LDS: 16x128 floats, float4 per thread x2
    for (int f = tid; f < 16 * 32; f += 256) {
        int row = f >> 5;
        int c4  = (f & 31) << 2;
        int r   = mbase + row; if (r >= nn) r = nn - 1;   // clamp (stores are guarded)
        float nrm = norm_src[r];
        const float4 v = *(const float4*)(feats + (size_t)r * 128 + c4);
        float4 s; s.x = v.x * nrm; s.y = v.y * nrm; s.z = v.z * nrm; s.w = v.w * nrm;
        *(float4*)(&As[row * LDSS + c4]) = s;
    }
    __syncthreads();

    const int col  = (wave << 4) + (lane & 15);
    const int krow = (lane >> 4) << 1;        // 0 for lanes 0-15, 2 for lanes 16-31
    v8f c = {};
    #pragma unroll 4
    for (int kk = 0; kk < 128; kk += 4) {
        const int k0 = kk + krow;
        v2f a = *(const v2f*)(&As[(lane & 15) * LDSS + k0]);   // A frag: M=lane%16, K=k0,k0+1
        v2f b;
        b.x = W1[(size_t)k0 * 128 + col];                      // B frag: K=k0/k0+1, N=col
        b.y = W1[(size_t)(k0 + 1) * 128 + col];
        c = __builtin_amdgcn_wmma_f32_16x16x4_f32(false, a, false, b, (short)0, c, false, false);
    }

    const int rbase = mbase + ((lane >> 4) << 3);              // lanes 16-31 hold M=8..15
    if (mbase + 16 <= nn) {
        // uniform fast path: one base pointer + immediate offsets, no exec-mask churn
        float* p = H1 + (size_t)rbase * 128 + col;
        #pragma unroll
        for (int v = 0; v < 8; ++v) p[v * 128] = c[v];
    } else {
        #pragma unroll
        for (int v = 0; v < 8; ++v) {
            int r = rbase + v;
            if (r < nn) H1[(size_t)r * 128 + col] = c[v];
        }
    }
}

// ---------------- edge scatter, 128 feats: agg1[dst] += H1[src]; 1 wave per edge ----------------

__global__ __launch_bounds__(256) void gcn_scatter128(const float* __restrict__ H,
                                                      const int* __restrict__ src,
                                                      const int* __restrict__ dst,
                                                      float* __restrict__ agg, int E) {
    int e = blockIdx.x * 8 + (threadIdx.x >> 5);
    if (e >= E) return;
    int lane = threadIdx.x & 31;
    int s = src[e], d = dst[e];
    const float4 m = *(const float4*)(H + (size_t)s * 128 + lane * 4);
    float* o = agg + (size_t)d * 128 + lane * 4;
    atomicAdd(o + 0, m.x); atomicAdd(o + 1, m.y);
    atomicAdd(o + 2, m.z); atomicAdd(o + 3, m.w);
}

// ---------------- layer 2 GEMM with fused relu(agg*nd + b1)*ns; [N,128]x[128,64] ----------------
// 128 threads = 4 waves; block handles 16 rows x 64 cols.

__global__ __launch_bounds__(128) void gcn_gemm2(const float* __restrict__ agg1,
                                                 const float* __restrict__ norm_src,
                                                 const float* __restrict__ norm_dst,
                                                 const float* __restrict__ b1,
                                                 const float* __restrict__ W2,
                                                 float* __restrict__ H2, int nn) {
    __shared__ float As[16 * LDSS];
    const int tid  = threadIdx.x;
    const int wave = tid >> 5;
    const int lane = tid & 31;
    const int mbase = blockIdx.x * 16;

    for (int f = tid; f < 16 * 32; f += 128) {
        int row = f >> 5;
        int c4  = (f & 31) << 2;
        int r   = mbase + row; if (r >= nn) r = nn - 1;
        float nd = norm_dst[r];
        float ns = norm_src[r];
        const float4 v  = *(const float4*)(agg1 + (size_t)r * 128 + c4);
        const float4 bb = *(const float4*)(b1 + c4);
        float4 s;
        s.x = fmaxf(v.x * nd + bb.x, 0.0f) * ns;
        s.y = fmaxf(v.y * nd + bb.y, 0.0f) * ns;
        s.z = fmaxf(v.z * nd + bb.z, 0.0f) * ns;
        s.w = fmaxf(v.w * nd + bb.w, 0.0f) * ns;
        *(float4*)(&As[row * LDSS + c4]) = s;
    }
    __syncthreads();

    const int col  = (wave << 4) + (lane & 15);
    const int krow = (lane >> 4) << 1;
    v8f c = {};
    #pragma unroll 4
    for (int kk = 0; kk < 128; kk += 4) {
        const int k0 = kk + krow;
        v2f a = *(const v2f*)(&As[(lane & 15) * LDSS + k0]);
        v2f b;
        b.x = W2[(size_t)k0 * 64 + col];
        b.y = W2[(size_t)(k0 + 1) * 64 + col];
        c = __builtin_amdgcn_wmma_f32_16x16x4_f32(false, a, false, b, (short)0, c, false, false);
    }

    const int rbase = mbase + ((lane >> 4) << 3);
    if (mbase + 16 <= nn) {
        float* p = H2 + (size_t)rbase * 64 + col;
        #pragma unroll
        for (int v = 0; v < 8; ++v) p[v * 64] = c[v];
    } else {
        #pragma unroll
        for (int v = 0; v < 8; ++v) {
            int r = rbase + v;
            if (r < nn) H2[(size_t)r * 64 + col] = c[v];
        }
    }
}

// ---------------- edge scatter, 64 feats ----------------

__global__ __launch_bounds__(256) void gcn_scatter64(const float* __restrict__ H,
                                                     const int* __restrict__ src,
                                                     const int* __restrict__ dst,
                                                     float* __restrict__ agg, int E) {
    int e = blockIdx.x * 8 + (threadIdx.x >> 5);
    if (e >= E) return;
    int lane = threadIdx.x & 31;
    int s = src[e], d = dst[e];
    const float2 m = *(const float2*)(H + (size_t)s * 64 + lane * 2);
    float* o = agg + (size_t)d * 64 + lane * 2;
    atomicAdd(o + 0, m.x); atomicAdd(o + 1, m.y);
}

// ---------------- finalize: out = agg2 * norm_dst + b2 ----------------

__global__ __launch_bounds__(256) void gcn_finalize(const float* __restrict__ agg2,
                                                    const float* __restrict__ norm_dst,
                                                    const float* __restrict__ b2,
                                                    float* __restrict__ out, long total) {
    long i = (long)blockIdx.x * blockDim.x + threadIdx.x;
    if (i < total) {
        long node = i >> 6;
        int  j    = (int)(i & 63);
        out[i] = agg2[i] * norm_dst[node] + b2[j];
    }
}

// ---------------- driver ----------------

extern "C" void kernel_launch(void* const* d_in, const int* in_sizes, int n_in,
                              void* d_out, int out_size, void* d_ws, size_t ws_size,
                              hipStream_t stream) {
    const float* feats = (const float*)d_in[0];   // [N,128]
    const float* W1    = (const float*)d_in[1];   // [128,128]
    const float* b1    = (const float*)d_in[2];   // [128]
    const float* W2    = (const float*)d_in[3];   // [128,64]
    const float* b2    = (const float*)d_in[4];   // [64]
    const int*   src   = (const int*)d_in[5];     // [E]
    const int*   dst   = (const int*)d_in[6];     // [E]

    const int N = in_sizes[0] / 128;
    const int E = in_sizes[5];

    // workspace layout (floats): [norm_src N][norm_dst N][agg1 N*128][agg2 N*64][H1 N*128][H2 N*64]
    float* ws       = (float*)d_ws;
    float* norm_src = ws;                  ws += N;
    float* norm_dst = ws;                  ws += N;
    float* agg1     = ws;                  ws += (size_t)N * 128;
    float* agg2     = ws;                  ws += (size_t)N * 64;
    float* H1       = ws;                  ws += (size_t)N * 128;
    float* H2       = ws;                  ws += (size_t)N * 64;

    // zero degrees + accumulators (first 194*N floats of workspace)
    const long nzero = (long)N * (2 + 128 + 64);
    gcn_zero<<<2048, 256, 0, stream>>>((float*)d_ws, nzero);

    // degrees -> norms (in place; norm_src/norm_dst are contiguous)
    gcn_degree<<<(E + 255) / 256, 256, 0, stream>>>(src, dst, norm_src, norm_dst, E);
    gcn_norm<<<(2 * N + 255) / 256, 256, 0, stream>>>((float*)d_ws, 2 * N);

    const int mtiles = (N + 15) / 16;

    // layer 1
    gcn_gemm1<<<mtiles, 256, 0, stream>>>(feats, norm_src, W1, H1, N);
    gcn_scatter128<<<(E + 7) / 8, 256, 0, stream>>>(H1, src, dst, agg1, E);

    // layer 2 (bias1 + relu + norms fused into A-load)
    gcn_gemm2<<<mtiles, 128, 0, stream>>>(agg1, norm_src, norm_dst, b1, W2, H2, N);
    gcn_scatter64<<<(E + 7) / 8, 256, 0, stream>>>(H2, src, dst, agg2, E);

    // out = agg2 * norm_dst + b2
    const long total = (long)N * 64;
    gcn_finalize<<<(int)((total + 255) / 256), 256, 0, stream>>>(agg2, norm_dst, b2,
                                                                 (float*)d_out, total);
}